// JobEmbedding_18468359373265
// MI455X (gfx1250) — compile-verified
//
#include <hip/hip_runtime.h>

// ---------------------------------------------------------------------------
// HGT/GAT job-embedding for MI455X (gfx1250, wave32).
// fp32 WMMA (V_WMMA_F32_16X16X4_F32) for the dense GEMMs; attention vectors
// algebraically folded (W_dst ∘ att_dst -> [128,4]) to kill the 5 big hd GEMMs;
// segment softmax via ordered-uint atomicMax + f32 atomic adds; final kernel
// fuses GEMM + residual + ReLU + LayerNorm using the WMMA D-matrix layout.
// Round 2: __launch_bounds__(128) on WMMA kernels to unlock the full VGPR
// file (kills the accumulator scratch spills seen in round 1 asm); NT hints
// on streamed agg/out traffic.
// ---------------------------------------------------------------------------

typedef __attribute__((ext_vector_type(2))) float v2f;
typedef __attribute__((ext_vector_type(8))) float v8f;

#define NJ    100000
#define DJ    128
#define OUTC  256
#define HH    4
#define CC    64
#define EDGES 500000
#define NEG_SLOPE 0.2f
#define ORD_NEG_INF 0x007FFFFFu   // ordered encoding of -inf

static __device__ __forceinline__ unsigned f2ord(float f) {
  unsigned u = __float_as_uint(f);
  return (u & 0x80000000u) ? ~u : (u | 0x80000000u);
}
static __device__ __forceinline__ float ord2f(unsigned u) {
  unsigned v = (u & 0x80000000u) ? (u & 0x7FFFFFFFu) : ~u;
  return __uint_as_float(v);
}
static __device__ __forceinline__ float lrelu(float x) {
  return x > 0.f ? x : NEG_SLOPE * x;
}

// ------------------------------- fills -------------------------------------
__global__ void k_fill_u32(unsigned* p, unsigned v, int n) {
  int i = blockIdx.x * blockDim.x + threadIdx.x;
  if (i < n) p[i] = v;
}
__global__ void k_fill_f32(float* p, float v, long n) {
  long i = (long)blockIdx.x * blockDim.x + threadIdx.x;
  if (i < n) p[i] = v;
}

// bias_total = sum of 5 relation biases (all folded into the final kernel)
__global__ void k_btot(const float* b0, const float* b1, const float* b2,
                       const float* b3, const float* b4, float* bt) {
  int i = threadIdx.x;  // 256 threads
  bt[i] = b0[i] + b1[i] + b2[i] + b3[i] + b4[i];
}

// vd[d, rel*4+h] = sum_c W_dst[d, h*64+c] * att_dst[h, c]   (fold hd GEMM away)
__global__ void k_vd(const float* __restrict__ Wd, const float* __restrict__ att,
                     float* __restrict__ vd, int rel) {
  int i = blockIdx.x * blockDim.x + threadIdx.x;
  if (i >= DJ * HH) return;
  int d = i >> 2, h = i & 3;
  float s = 0.f;
  for (int c = 0; c < CC; ++c) s += Wd[d * OUTC + h * CC + c] * att[h * CC + c];
  vd[d * 20 + rel * 4 + h] = s;
}

// a_d_all[j, 0..19] = x_job[j] @ vd  (all 5 relations' dst-attention at once)
__global__ void k_ad(const float* __restrict__ xj, const float* __restrict__ vd,
                     float* __restrict__ ad) {
  __shared__ float sv[DJ * 20];
  for (int i = threadIdx.x; i < DJ * 20; i += blockDim.x) sv[i] = vd[i];
  __syncthreads();
  int j = blockIdx.x * blockDim.x + threadIdx.x;
  if (j >= NJ) return;
  float acc[20];
#pragma unroll
  for (int t = 0; t < 20; ++t) acc[t] = 0.f;
  const float* xr = xj + (long)j * DJ;
  for (int d = 0; d < DJ; ++d) {
    float x = xr[d];
#pragma unroll
    for (int t = 0; t < 20; ++t) acc[t] += x * sv[d * 20 + t];
  }
#pragma unroll
  for (int t = 0; t < 20; ++t) ad[(long)j * 20 + t] = acc[t];
}

// ---------------- fp32 WMMA GEMM: C[M,256] = A[M,K] @ B[K,256] --------------
// One wave computes a 16x256 strip (16 N-tiles x 8 acc VGPRs = 128 VGPRs).
// __launch_bounds__(128): 4 waves/block -> 1 wave/SIMD -> full VGPR budget,
// no accumulator spills.
__global__ __launch_bounds__(128) void k_gemm16(
    const float* __restrict__ A, int M, int K,
    const float* __restrict__ B, float* __restrict__ C) {
  int wave = (blockIdx.x * blockDim.x + threadIdx.x) >> 5;
  int m0 = wave * 16;
  if (m0 >= M) return;                     // wave-uniform: EXEC stays all-ones
  int lane = threadIdx.x & 31;
  int g = lane >> 4, lm = lane & 15;
  v8f acc[16] = {};
  for (int k = 0; k < K; k += 4) {
    const float* ap = A + (long)(m0 + lm) * K + k + 2 * g;
    v2f a; a.x = ap[0]; a.y = ap[1];
#pragma unroll
    for (int nt = 0; nt < 16; ++nt) {
      const float* bp = B + (long)(k + 2 * g) * OUTC + nt * 16 + lm;
      v2f b; b.x = bp[0]; b.y = bp[OUTC];
      acc[nt] = __builtin_amdgcn_wmma_f32_16x16x4_f32(
          false, a, false, b, (short)0, acc[nt], false, false);
    }
  }
#pragma unroll
  for (int nt = 0; nt < 16; ++nt)
#pragma unroll
    for (int r = 0; r < 8; ++r)
      C[(long)(m0 + r + 8 * g) * OUTC + nt * 16 + lm] = acc[nt][r];
}

// a_s[n,h] = sum_c hs[n, h*64+c] * att_src[h,c]
__global__ void k_as(const float* __restrict__ hs, const float* __restrict__ att,
                     float* __restrict__ as_, int n_src) {
  int i = blockIdx.x * blockDim.x + threadIdx.x;
  if (i >= n_src * HH) return;
  int n = i >> 2, h = i & 3;
  const float* hr = hs + (long)n * OUTC + h * CC;
  float s = 0.f;
  for (int c = 0; c < CC; ++c) s += hr[c] * att[h * CC + c];
  as_[i] = s;
}

// pass 1: segment max of leaky-relu logits (ordered-uint atomicMax)
__global__ void k_edge_max(const int* __restrict__ src, const int* __restrict__ dst,
                           const float* __restrict__ as_, const float* __restrict__ ad,
                           int rel, unsigned* __restrict__ m_ord) {
  int e = blockIdx.x * blockDim.x + threadIdx.x;
  if (e >= EDGES) return;
  int s = src[e], d = dst[e];
  float4 a4 = *(const float4*)(as_ + (long)s * 4);
  float4 d4 = *(const float4*)(ad + (long)d * 20 + rel * 4);
  unsigned* mp = m_ord + (long)d * 4;
  atomicMax(&mp[0], f2ord(lrelu(a4.x + d4.x)));
  atomicMax(&mp[1], f2ord(lrelu(a4.y + d4.y)));
  atomicMax(&mp[2], f2ord(lrelu(a4.z + d4.z)));
  atomicMax(&mp[3], f2ord(lrelu(a4.w + d4.w)));
}

// pass 2: ex = exp(l - m), stash per-edge, atomic-add segment denominators
__global__ void k_edge_exp(const int* __restrict__ src, const int* __restrict__ dst,
                           const float* __restrict__ as_, const float* __restrict__ ad,
                           int rel, const unsigned* __restrict__ m_ord,
                           float* __restrict__ denom, float* __restrict__ ex_e) {
  int e = blockIdx.x * blockDim.x + threadIdx.x;
  if (e >= EDGES) return;
  int s = src[e], d = dst[e];
  float4 a4 = *(const float4*)(as_ + (long)s * 4);
  float4 d4 = *(const float4*)(ad + (long)d * 20 + rel * 4);
  const unsigned* mp = m_ord + (long)d * 4;
  float4 ex;
  ex.x = expf(lrelu(a4.x + d4.x) - ord2f(mp[0]));
  ex.y = expf(lrelu(a4.y + d4.y) - ord2f(mp[1]));
  ex.z = expf(lrelu(a4.z + d4.z) - ord2f(mp[2]));
  ex.w = expf(lrelu(a4.w + d4.w) - ord2f(mp[3]));
  *(float4*)(ex_e + (long)e * 4) = ex;
  float* dn = denom + (long)d * 4;
  __hip_atomic_fetch_add(&dn[0], ex.x, __ATOMIC_RELAXED, __HIP_MEMORY_SCOPE_AGENT);
  __hip_atomic_fetch_add(&dn[1], ex.y, __ATOMIC_RELAXED, __HIP_MEMORY_SCOPE_AGENT);
  __hip_atomic_fetch_add(&dn[2], ex.z, __ATOMIC_RELAXED, __HIP_MEMORY_SCOPE_AGENT);
  __hip_atomic_fetch_add(&dn[3], ex.w, __ATOMIC_RELAXED, __HIP_MEMORY_SCOPE_AGENT);
}

// pass 3: one wave per edge; agg[dst] += hs[src] * alpha (coalesced f32 atomics)
__global__ void k_edge_scatter(const int* __restrict__ src, const int* __restrict__ dst,
                               const float* __restrict__ ex_e, const float* __restrict__ denom,
                               const float* __restrict__ hs, float* __restrict__ agg) {
  int wave = (blockIdx.x * blockDim.x + threadIdx.x) >> 5;
  if (wave >= EDGES) return;
  int lane = threadIdx.x & 31;
  int s = src[wave], d = dst[wave];
  float4 ex4 = *(const float4*)(ex_e + (long)wave * 4);
  float4 dn4 = *(const float4*)(denom + (long)d * 4);
  float alpha[4] = { ex4.x / (dn4.x + 1e-16f), ex4.y / (dn4.y + 1e-16f),
                     ex4.z / (dn4.z + 1e-16f), ex4.w / (dn4.w + 1e-16f) };
  const float* hrow = hs + (long)s * OUTC;
  float* arow = agg + (long)d * OUTC;
#pragma unroll
  for (int k = 0; k < 8; ++k) {
    int ch = lane + 32 * k;                 // head = k>>1 is wave-uniform
    __hip_atomic_fetch_add(&arow[ch], hrow[ch] * alpha[k >> 1],
                           __ATOMIC_RELAXED, __HIP_MEMORY_SCOPE_AGENT);
  }
}

// final: out = LN(relu(agg + bias_total + x_job @ W_res)); WMMA GEMM fused
// with LayerNorm — each output row lives in one 16-lane half-group of D.
__global__ __launch_bounds__(128) void k_final(
    const float* __restrict__ xj, const float* __restrict__ Wres,
    const float* __restrict__ agg, const float* __restrict__ btot,
    const float* __restrict__ gam, const float* __restrict__ bet,
    float* __restrict__ out) {
  int wave = (blockIdx.x * blockDim.x + threadIdx.x) >> 5;
  int m0 = wave * 16;
  if (m0 >= NJ) return;
  int lane = threadIdx.x & 31;
  int g = lane >> 4, lm = lane & 15;
  v8f acc[16] = {};
  for (int k = 0; k < DJ; k += 4) {
    const float* ap = xj + (long)(m0 + lm) * DJ + k + 2 * g;
    v2f a; a.x = ap[0]; a.y = ap[1];
#pragma unroll
    for (int nt = 0; nt < 16; ++nt) {
      const float* bp = Wres + (long)(k + 2 * g) * OUTC + nt * 16 + lm;
      v2f b; b.x = bp[0]; b.y = bp[OUTC];
      acc[nt] = __builtin_amdgcn_wmma_f32_16x16x4_f32(
          false, a, false, b, (short)0, acc[nt], false, false);
    }
  }
  // + agg (streamed once: NT load) + bias, ReLU
#pragma unroll
  for (int nt = 0; nt < 16; ++nt) {
    int col = nt * 16 + lm;
    float bt = btot[col];
#pragma unroll
    for (int r = 0; r < 8; ++r) {
      long m = m0 + r + 8 * g;
      float v = acc[nt][r] + __builtin_nontemporal_load(&agg[m * OUTC + col]) + bt;
      acc[nt][r] = fmaxf(v, 0.f);
    }
  }
  // per-row LayerNorm: row owned by one 16-lane half-group across nt regs
#pragma unroll
  for (int r = 0; r < 8; ++r) {
    float s = 0.f;
#pragma unroll
    for (int nt = 0; nt < 16; ++nt) s += acc[nt][r];
#pragma unroll
    for (int msk = 1; msk < 16; msk <<= 1) s += __shfl_xor(s, msk, 32);
    float mu = s * (1.f / OUTC);
    float s2 = 0.f;
#pragma unroll
    for (int nt = 0; nt < 16; ++nt) { float dv = acc[nt][r] - mu; s2 += dv * dv; }
#pragma unroll
    for (int msk = 1; msk < 16; msk <<= 1) s2 += __shfl_xor(s2, msk, 32);
    float rstd = rsqrtf(s2 * (1.f / OUTC) + 1e-5f);
    long m = m0 + r + 8 * g;
#pragma unroll
    for (int nt = 0; nt < 16; ++nt) {
      int col = nt * 16 + lm;
      float o = gam[col] * (acc[nt][r] - mu) * rstd + bet[col];
      __builtin_nontemporal_store(o, &out[m * OUTC + col]);
    }
  }
}

// ---------------------------------------------------------------------------
extern "C" void kernel_launch(void* const* d_in, const int* in_sizes, int n_in,
                              void* d_out, int out_size, void* d_ws, size_t ws_size,
                              hipStream_t stream) {
  (void)in_sizes; (void)n_in; (void)out_size; (void)ws_size;
  const float* x_job = (const float*)d_in[0];
  const float* x_station = (const float*)d_in[1];
  const float* x_machine = (const float*)d_in[2];
  const float* x_robot = (const float*)d_in[3];
  const float* W_res = (const float*)d_in[39];
  const float* ln_g = (const float*)d_in[40];
  const float* ln_b = (const float*)d_in[41];

  struct RelInfo { const float* xs; int n_src; int d_src; };
  RelInfo rels[5] = { { x_station, 2048, 64 }, { x_station, 2048, 64 },
                      { x_machine, 512, 64 },  { x_machine, 512, 64 },
                      { x_robot, 256, 32 } };

  // workspace carve-up (floats); ~124 MB total
  float* ws = (float*)d_ws;
  float* agg = ws;                                  // NJ*256
  float* ad = agg + (long)NJ * OUTC;                // NJ*20
  float* hs = ad + (long)NJ * 20;                   // 2048*256 (reused per rel)
  float* as_ = hs + (long)2048 * OUTC;              // 2048*4
  unsigned* m_ord = (unsigned*)(as_ + 2048 * 4);    // NJ*4
  float* denom = (float*)(m_ord + (long)NJ * 4);    // NJ*4
  float* ex_e = denom + (long)NJ * 4;               // E*4
  float* vd = ex_e + (long)EDGES * 4;               // 128*20
  float* btot = vd + 128 * 20;                      // 256

  const int T = 256;
  const int TW = 128;  // WMMA kernels: 4 waves/block, full VGPR budget
  long nagg = (long)NJ * OUTC;
  k_fill_f32<<<(unsigned)((nagg + T - 1) / T), T, 0, stream>>>(agg, 0.f, nagg);
  k_btot<<<1, 256, 0, stream>>>((const float*)d_in[10], (const float*)d_in[17],
                                (const float*)d_in[24], (const float*)d_in[31],
                                (const float*)d_in[38], btot);
  for (int r = 0; r < 5; ++r)
    k_vd<<<(DJ * HH + T - 1) / T, T, 0, stream>>>(
        (const float*)d_in[7 + 7 * r], (const float*)d_in[9 + 7 * r], vd, r);
  k_ad<<<(NJ + T - 1) / T, T, 0, stream>>>(x_job, vd, ad);

  for (int r = 0; r < 5; ++r) {
    const int* srcp = (const int*)d_in[4 + 7 * r];
    const int* dstp = (const int*)d_in[5 + 7 * r];
    const float* Wsrc = (const float*)d_in[6 + 7 * r];
    const float* att_s = (const float*)d_in[8 + 7 * r];
    int M = rels[r].n_src, K = rels[r].d_src;

    k_fill_u32<<<(NJ * 4 + T - 1) / T, T, 0, stream>>>(m_ord, ORD_NEG_INF, NJ * 4);
    k_fill_f32<<<(NJ * 4 + T - 1) / T, T, 0, stream>>>(denom, 0.f, (long)NJ * 4);

    int waves = M / 16;
    k_gemm16<<<(waves * 32 + TW - 1) / TW, TW, 0, stream>>>(rels[r].xs, M, K, Wsrc, hs);
    k_as<<<(M * HH + T - 1) / T, T, 0, stream>>>(hs, att_s, as_, M);

    k_edge_max<<<(EDGES + T - 1) / T, T, 0, stream>>>(srcp, dstp, as_, ad, r, m_ord);
    k_edge_exp<<<(EDGES + T - 1) / T, T, 0, stream>>>(srcp, dstp, as_, ad, r, m_ord,
                                                      denom, ex_e);
    k_edge_scatter<<<(unsigned)(((long)EDGES * 32 + T - 1) / T), T, 0, stream>>>(
        srcp, dstp, ex_e, denom, hs, agg);
  }

  int fwaves = NJ / 16;  // 6250
  k_final<<<(fwaves * 32 + TW - 1) / TW, TW, 0, stream>>>(x_job, W_res, agg, btot,
                                                          ln_g, ln_b, (float*)d_out);
}